// HybridProjection3D_54073638256834
// MI455X (gfx1250) — compile-verified
//
#include <hip/hip_runtime.h>

typedef __attribute__((ext_vector_type(2))) float v2f;
typedef __attribute__((ext_vector_type(8))) float v8f;

#define C_INW 8
#define C_OUTW 8
#define NN 512      // nodes = 8*8*8
#define DD 4096     // C_IN * NN
#define RR 8
#define NTAP 15     // 7 stencil + 8 random
#define NSTEP 30    // K=120 / 4

__global__ __launch_bounds__(256) void hybrid3d_wmma_kernel(
    const float* __restrict__ x,        // (2048, 4096)  sample-major, c*512+n
    const float* __restrict__ conv_w,   // (8,8,27)
    const float* __restrict__ conv_b,   // (8)
    const float* __restrict__ mix_w,    // (8,8,8)
    const float* __restrict__ mix_b,    // (8)
    const float* __restrict__ alpha_p,  // (1)
    const int*   __restrict__ rand_idx, // (512,8)
    float* __restrict__ out)            // (2048, 4096)  o*512+n
{
  const int lane   = threadIdx.x & 31;
  const int wave   = threadIdx.x >> 5;
  const int lane_m = lane & 15;   // A row (sample) / B,D col (out channel)
  const int khalf  = lane >> 4;   // selects K pair {2*khalf, 2*khalf+1}

  const int sblk  = blockIdx.x;                 // 0..127 : 16 samples each
  const int node0 = blockIdx.y * 64 + wave * 8; // 8 nodes per wave

  const float alpha   = alpha_p[0];
  const float a       = 1.0f / (1.0f + __expf(-alpha));
  const float one_m_a = 1.0f - a;

  // stencil tap order: center, -x, +x, -y, +y, -z, +z  (3x3x3 flat indices)
  const int widx[7] = {13, 4, 22, 10, 16, 12, 14};

  // ---- B fragments: pre-blended weights, built once, reused for all nodes ----
  // step s = 2*t + j covers k = 4s..4s+3 -> channels c = 4j + 2*khalf + {0,1} of tap t
  v2f bfrag[NSTEP];
  #pragma unroll
  for (int t = 0; t < NTAP; ++t) {
    #pragma unroll
    for (int j = 0; j < 2; ++j) {
      const int c0 = 4 * j + 2 * khalf;
      float w0 = 0.f, w1 = 0.f;
      if (lane_m < C_OUTW) {
        if (t < 7) {
          w0 = a * conv_w[(lane_m * C_INW + c0    ) * 27 + widx[t]];
          w1 = a * conv_w[(lane_m * C_INW + c0 + 1) * 27 + widx[t]];
        } else {
          w0 = one_m_a * mix_w[(lane_m * C_INW + c0    ) * RR + (t - 7)];
          w1 = one_m_a * mix_w[(lane_m * C_INW + c0 + 1) * RR + (t - 7)];
        }
      }
      v2f b; b.x = w0; b.y = w1;
      bfrag[2 * t + j] = b;
    }
  }

  float bias = 0.f;
  if (lane_m < C_OUTW) bias = a * conv_b[lane_m] + one_m_a * mix_b[lane_m];

  // Scalar (wave-uniform) base for this sample block; per-lane 32-bit offset.
  const float*   xs   = x + (size_t)(sblk * 16) * DD;
  const unsigned voff = (unsigned)(lane_m * DD + khalf * (2 * NN)); // elements

  #pragma unroll 1
  for (int np = 0; np < 4; ++np) {     // 2 adjacent nodes per iteration
    // Force the compiler to treat node ids (and everything derived) as scalar.
    const int nA = __builtin_amdgcn_readfirstlane(node0 + 2 * np);
    const int nB = nA + 1;

    unsigned sgA[NTAP], sgB[NTAP];     // unsigned: no sign-extend SALU chains
    float    mA[6], mB[6];             // 0/1 masks for the 6 boundary taps
    {
      int n = nA;
      int nx = n >> 6, ny = (n >> 3) & 7, nz = n & 7;
      sgA[0] = (unsigned)n;
      sgA[1] = (unsigned)((nx > 0) ? n - 64 : n);  mA[0] = (nx > 0) ? 1.f : 0.f;
      sgA[2] = (unsigned)((nx < 7) ? n + 64 : n);  mA[1] = (nx < 7) ? 1.f : 0.f;
      sgA[3] = (unsigned)((ny > 0) ? n - 8  : n);  mA[2] = (ny > 0) ? 1.f : 0.f;
      sgA[4] = (unsigned)((ny < 7) ? n + 8  : n);  mA[3] = (ny < 7) ? 1.f : 0.f;
      sgA[5] = (unsigned)((nz > 0) ? n - 1  : n);  mA[4] = (nz > 0) ? 1.f : 0.f;
      sgA[6] = (unsigned)((nz < 7) ? n + 1  : n);  mA[5] = (nz < 7) ? 1.f : 0.f;
      #pragma unroll
      for (int r = 0; r < RR; ++r) sgA[7 + r] = (unsigned)rand_idx[n * RR + r];

      n = nB;
      nx = n >> 6; ny = (n >> 3) & 7; nz = n & 7;
      sgB[0] = (unsigned)n;
      sgB[1] = (unsigned)((nx > 0) ? n - 64 : n);  mB[0] = (nx > 0) ? 1.f : 0.f;
      sgB[2] = (unsigned)((nx < 7) ? n + 64 : n);  mB[1] = (nx < 7) ? 1.f : 0.f;
      sgB[3] = (unsigned)((ny > 0) ? n - 8  : n);  mB[2] = (ny > 0) ? 1.f : 0.f;
      sgB[4] = (unsigned)((ny < 7) ? n + 8  : n);  mB[3] = (ny < 7) ? 1.f : 0.f;
      sgB[5] = (unsigned)((nz > 0) ? n - 1  : n);  mB[4] = (nz > 0) ? 1.f : 0.f;
      sgB[6] = (unsigned)((nz < 7) ? n + 1  : n);  mB[5] = (nz < 7) ? 1.f : 0.f;
      #pragma unroll
      for (int r = 0; r < RR; ++r) sgB[7 + r] = (unsigned)rand_idx[n * RR + r];
    }

    v8f accA = {}; v8f accB = {};
    #pragma unroll
    for (int t = 0; t < NTAP; ++t) {
      // Scalar base per tap; 4 immediate-offset loads per node cover
      // (j,e) in {0,1}x{0,1}:  element offsets 0, 512, 2048, 2560.
      const float* pA = xs + sgA[t];
      const float* pB = xs + sgB[t];
      float a0 = pA[voff];        float b0 = pB[voff];
      float a1 = pA[voff + 512];  float b1 = pB[voff + 512];
      float a2 = pA[voff + 2048]; float b2 = pB[voff + 2048];
      float a3 = pA[voff + 2560]; float b3 = pB[voff + 2560];
      if (t >= 1 && t < 7) {      // branchless zero for OOB stencil neighbors
        const float ma = mA[t - 1], mb = mB[t - 1];
        a0 *= ma; a1 *= ma; a2 *= ma; a3 *= ma;
        b0 *= mb; b1 *= mb; b2 *= mb; b3 *= mb;
      }
      v2f A0; A0.x = a0; A0.y = a1;
      v2f A1; A1.x = a2; A1.y = a3;
      v2f B0; B0.x = b0; B0.y = b1;
      v2f B1; B1.x = b2; B1.y = b3;
      accA = __builtin_amdgcn_wmma_f32_16x16x4_f32(
                 false, A0, false, bfrag[2 * t    ], (short)0, accA, false, false);
      accB = __builtin_amdgcn_wmma_f32_16x16x4_f32(
                 false, B0, false, bfrag[2 * t    ], (short)0, accB, false, false);
      accA = __builtin_amdgcn_wmma_f32_16x16x4_f32(
                 false, A1, false, bfrag[2 * t + 1], (short)0, accA, false, false);
      accB = __builtin_amdgcn_wmma_f32_16x16x4_f32(
                 false, B1, false, bfrag[2 * t + 1], (short)0, accB, false, false);
    }

    // D layout: lane col = lane_m (= out channel), VGPR r -> sample row r + 8*khalf.
    // nA, nB are adjacent in memory -> pack into one aligned b64 store each row.
    if (lane_m < C_OUTW) {
      #pragma unroll
      for (int r = 0; r < 8; ++r) {
        const int samp = r + 8 * khalf;
        const size_t o0 = (size_t)(sblk * 16 + samp) * DD + (size_t)lane_m * NN;
        v2f v; v.x = accA[r] + bias; v.y = accB[r] + bias;
        *(v2f*)(out + o0 + (unsigned)nA) = v;   // 8B-aligned: nA even, o0 % 512 == 0
      }
    }
  }
}

extern "C" void kernel_launch(void* const* d_in, const int* in_sizes, int n_in,
                              void* d_out, int out_size, void* d_ws, size_t ws_size,
                              hipStream_t stream) {
  const float* x        = (const float*)d_in[0];
  const float* conv_w   = (const float*)d_in[1];
  const float* conv_b   = (const float*)d_in[2];
  const float* mix_w    = (const float*)d_in[3];
  const float* mix_b    = (const float*)d_in[4];
  const float* alpha    = (const float*)d_in[5];
  const int*   rand_idx = (const int*)d_in[6];
  float* out = (float*)d_out;

  // 128 sample-blocks (16 samples) x 8 node-groups (64 nodes); 8 waves/block
  dim3 grid(128, 8, 1);
  dim3 block(256, 1, 1);
  hybrid3d_wmma_kernel<<<grid, block, 0, stream>>>(
      x, conv_w, conv_b, mix_w, mix_b, alpha, rand_idx, out);
}